// ContinuousKoopman_86629490361064
// MI455X (gfx1250) — compile-verified
//
#include <hip/hip_runtime.h>
#include <hip/hip_bf16.h>

// ---------------------------------------------------------------------------
// pred[b] = x[b] @ expm(dt[b] * K),  K = tridiag(-off, -diag^2, +off), N=64.
// expm(dt*K) = sum_k dt^k * (K^k / k!)  ->  Y = X_cat @ M_cat  (one GEMM),
// with X_cat[b, 64k+n] = dt[b]^k * x[b,n] built in registers.
// T=12 Taylor terms: ||dt*K|| <~ 0.5  =>  remainder ~5e-13 (exact in fp32).
//
// Phase 1: build M_k = K^k/k! (tridiagonal recurrence, 3 FMAs/element) into
//          d_ws in a PAIR-INTERLEAVED layout so phase 2's B fragments are
//          single aligned b64 loads:
//            P[k][rp][n][s] = M_k[2*rp + s][n]   (rp=0..31, n=0..63, s=0..1)
// Phase 2: wave32 tiles, V_WMMA_F32_16X16X4_F32 (full-fp32 tensor op) GEMM.
//          c-loop fully unrolled -> static VGPR indexing (no v_movrels/M0).
// Problem is HBM-bound (~4 MB mandatory traffic -> ~0.2us @ 23.3 TB/s);
// M_cat is 192 KB and stays resident in the 192 MB L2 across all 512 waves.
// ---------------------------------------------------------------------------

#define KOOP_N      64
#define KOOP_TERMS  12
#define KOOP_MATSZ  (KOOP_N * KOOP_N)   // 4096 floats per term matrix

typedef float v2f __attribute__((ext_vector_type(2)));
typedef float v8f __attribute__((ext_vector_type(8)));

// ---------------------------------------------------------------------------
// Phase 1: one workgroup (64 threads) builds M_k = K^k / k!, k = 0..T-1.
// (M @ K)[i][j] = M[i][j-1]*off[j-1] + M[i][j]*(-d_j^2) - M[i][j+1]*off[j].
// Output (pair-interleaved): Pcat[k*4096 + (i>>1)*128 + j*2 + (i&1)].
// ---------------------------------------------------------------------------
__global__ __launch_bounds__(64)
void koopman_build_terms(const float* __restrict__ diag,
                         const float* __restrict__ off_diags,
                         float* __restrict__ Pcat) {
    __shared__ float Ma[KOOP_MATSZ];
    __shared__ float Mb[KOOP_MATSZ];
    __shared__ float Kd[KOOP_N];   // main diagonal: -diag^2
    __shared__ float Ko[KOOP_N];   // off_diags, padded with 0 at index 63

    const int t = threadIdx.x;     // 0..63 : column owned by this thread

    {
        float d = diag[t];
        Kd[t] = -d * d;
        Ko[t] = (t < KOOP_N - 1) ? off_diags[t] : 0.0f;
    }
    __syncthreads();

    // M_0 = I
    float* cur = Ma;
    float* nxt = Mb;
    for (int i = 0; i < KOOP_N; ++i) {
        float v = (i == t) ? 1.0f : 0.0f;
        cur[i * KOOP_N + t] = v;
        Pcat[(i >> 1) * 128 + t * 2 + (i & 1)] = v;
    }
    __syncthreads();

    for (int k = 1; k < KOOP_TERMS; ++k) {
        const float inv = 1.0f / (float)k;
        float* __restrict__ dst = Pcat + k * KOOP_MATSZ;
        for (int i = 0; i < KOOP_N; ++i) {
            const float* row = cur + i * KOOP_N;
            float acc = row[t] * Kd[t];
            if (t > 0)            acc += row[t - 1] * Ko[t - 1];
            if (t < KOOP_N - 1)   acc -= row[t + 1] * Ko[t];
            float v = acc * inv;
            nxt[i * KOOP_N + t] = v;
            dst[(i >> 1) * 128 + t * 2 + (i & 1)] = v;
        }
        __syncthreads();
        float* tmp = cur; cur = nxt; nxt = tmp;
    }
}

// ---------------------------------------------------------------------------
// Phase 2: one wave (32 lanes) per 16-row tile of x.
//   acc_j (16x16 f32, j=0..3 covering output cols 16j..16j+15)
//     += A(16x4) x B(4x16) over all 4-wide K-chunks c and Taylor terms k,
//        where A = diag(dt^k) * X[:, 4c..4c+3].
// A-layout (32-bit A 16x4): lanes 0-15 hold K=0,1 ; lanes 16-31 hold K=2,3,
//   row M = lane&15.
// B-layout (4x16): VGPR0 = rows K0 (lanes 0-15) / K2 (lanes 16-31),
//                  VGPR1 = rows K1 / K3, col N = lane&15 (+16j).
//   -> pair {row 2rp, row 2rp+1} at column n is ONE b64 from Pcat.
// C/D layout (16x16 f32): VGPR v, lanes 0-15 -> M=v, lanes 16-31 -> M=v+8.
// ---------------------------------------------------------------------------
__global__ __launch_bounds__(32)
void koopman_apply(const float* __restrict__ x,
                   const float* __restrict__ dt,
                   const float* __restrict__ Pcat,
                   float* __restrict__ out,
                   int B) {
    const int row0 = blockIdx.x * 16;
    if (row0 >= B) return;

    const int lane = threadIdx.x;      // 0..31, EXEC all-ones (WMMA requirement)
    const int r    = lane & 15;        // A-row / B-column index for this lane
    const int hi   = lane >> 4;        // 0: K-pair {0,1}; 1: K-pair {2,3}

    // Preload this lane's half of x-row `row0+r`: for chunk c it needs the
    // adjacent pair (4c+2*hi, 4c+2*hi+1) -> one aligned b64 per chunk.
    v2f xr[16];
    const v2f* __restrict__ xrow =
        (const v2f*)(x + (size_t)(row0 + r) * KOOP_N + 2 * hi);
    #pragma unroll
    for (int c = 0; c < 16; ++c) {
        xr[c] = xrow[2 * c];           // floats 4c+2hi, 4c+2hi+1
    }
    const float dtv = dt[row0 + r];

    v8f acc0 = {}, acc1 = {}, acc2 = {}, acc3 = {};
    float p = 1.0f;                    // dt^k for this lane's row

    for (int k = 0; k < KOOP_TERMS; ++k) {
        const v2f* __restrict__ Pk = (const v2f*)(Pcat + k * KOOP_MATSZ);
        if (k + 1 < KOOP_TERMS) {
            // pull next term matrix toward L2/L0 (global_prefetch_b8)
            __builtin_prefetch(Pcat + (k + 1) * KOOP_MATSZ + (lane << 7), 0, 0);
        }
        #pragma unroll
        for (int c = 0; c < 16; ++c) {
            v2f a = { xr[c].x * p, xr[c].y * p };
            // row-pair rp = 2c+hi covers rows {4c+2hi, 4c+2hi+1}
            const v2f* __restrict__ bp = Pk + (2 * c + hi) * KOOP_N + r;
            v2f b0 = bp[0];
            v2f b1 = bp[16];
            v2f b2 = bp[32];
            v2f b3 = bp[48];
            acc0 = __builtin_amdgcn_wmma_f32_16x16x4_f32(false, a, false, b0,
                                                         (short)0, acc0, false, false);
            acc1 = __builtin_amdgcn_wmma_f32_16x16x4_f32(false, a, false, b1,
                                                         (short)0, acc1, false, false);
            acc2 = __builtin_amdgcn_wmma_f32_16x16x4_f32(false, a, false, b2,
                                                         (short)0, acc2, false, false);
            acc3 = __builtin_amdgcn_wmma_f32_16x16x4_f32(false, a, false, b3,
                                                         (short)0, acc3, false, false);
        }
        p *= dtv;
    }

    // Scatter C/D layout to out[row][col]: VGPR v of lane -> row0 + v + 8*hi.
    float* __restrict__ obase = out + (size_t)(row0 + 8 * hi) * KOOP_N + r;
    #pragma unroll
    for (int v = 0; v < 8; ++v) {
        float* __restrict__ orow = obase + (size_t)v * KOOP_N;
        orow[0]  = acc0[v];
        orow[16] = acc1[v];
        orow[32] = acc2[v];
        orow[48] = acc3[v];
    }
}

// ---------------------------------------------------------------------------
// Launch: inputs in setup_inputs() order: x (B*64), dt (B), diag (64),
// off_diags (63). Workspace needs T*64*64*4 = 196,608 bytes for M_cat.
// ---------------------------------------------------------------------------
extern "C" void kernel_launch(void* const* d_in, const int* in_sizes, int n_in,
                              void* d_out, int out_size, void* d_ws, size_t ws_size,
                              hipStream_t stream) {
    const float* x    = (const float*)d_in[0];
    const float* dt   = (const float*)d_in[1];
    const float* diag = (const float*)d_in[2];
    const float* offd = (const float*)d_in[3];
    float* out  = (float*)d_out;
    float* Pcat = (float*)d_ws;   // 12 * 16 KB = 192 KB of scratch

    const int B = in_sizes[0] / KOOP_N;   // 8192

    koopman_build_terms<<<1, 64, 0, stream>>>(diag, offd, Pcat);

    const int tiles = (B + 15) / 16;      // 512 single-wave workgroups
    koopman_apply<<<tiles, 32, 0, stream>>>(x, dt, Pcat, out, B);
}